// BilateralRotation_85263690760273
// MI455X (gfx1250) — compile-verified
//
#include <hip/hip_runtime.h>

#define NUM_HEADS 32
#define BATCH 512
#define LDSTR 68   // padded LDS row stride (floats): 68 % 64 == 4 -> conflict-free, 16B-aligned

typedef __attribute__((ext_vector_type(2))) float v2f;
typedef __attribute__((ext_vector_type(8))) float v8f;
typedef __attribute__((ext_vector_type(4))) int   v4i;

// --------------------------- async staging helpers -------------------------
#if __has_builtin(__builtin_amdgcn_global_load_async_to_lds_b128)
#define HAVE_ASYNC_LDS 1
#endif

// Builtin signature (from hipcc diagnostic): param0 = v4i in AS1 (__device__),
// param1 = v4i in AS3 (__shared__), then constant offset + cpol.
#define GPTR(p) ((__attribute__((address_space(1))) v4i*)(unsigned long long)(p))
#define LPTR(p) ((__attribute__((address_space(3))) v4i*)(unsigned int)(unsigned long long)(p))

__device__ __forceinline__ void stage16(const float* __restrict__ g, float* l) {
#ifdef HAVE_ASYNC_LDS
  // direct global -> LDS, 16B per lane, tracked by ASYNCcnt (no VGPR round-trip)
  __builtin_amdgcn_global_load_async_to_lds_b128(GPTR(g), LPTR(l), 0, 0);
#else
  *(float4*)l = *(const float4*)g;
#endif
}

__device__ __forceinline__ void stage_join() {
#ifdef HAVE_ASYNC_LDS
#if __has_builtin(__builtin_amdgcn_s_wait_asynccnt)
  __builtin_amdgcn_s_wait_asynccnt(0);
#else
  asm volatile("s_wait_asynccnt 0" ::: "memory");
#endif
#endif
}

// ---------------------------------------------------------------------------
// Kernel 1: Cayley transform.  Solve (I+A) R = (I-A), A = 0.5(p - p^T).
// One 256-thread block per 64x64 matrix.  blockIdx.x in [0,64):
//   z <  32 : p_left  head z    -> R1 slot z      (stored row-major, GEMM2 A)
//   z >= 32 : p_right head z-32 -> R2 slot z      (stored TRANSPOSED, GEMM1 B)
// Gauss-Jordan without pivoting (I+A has eigvals 1+i*lambda -> well-posed).
// ---------------------------------------------------------------------------
__global__ __launch_bounds__(256) void cayley_kernel(const float* __restrict__ p_left,
                                                     const float* __restrict__ p_right,
                                                     float* __restrict__ R) {
  __shared__ float Aug[64 * 132];   // 64 rows x 128 cols, padded to 132
  __shared__ float s_inv;
  const int z = blockIdx.x;
  const float* p = (z < NUM_HEADS) ? (p_left + z * 4096)
                                   : (p_right + (z - NUM_HEADS) * 4096);
  const int tid = threadIdx.x;
  const int row = tid >> 2;   // 4 threads per row
  const int sub = tid & 3;

  // Build augmented [I+A | I-A]
  for (int jj = 0; jj < 16; ++jj) {
    int j = sub * 16 + jj;
    float s = 0.5f * (p[row * 64 + j] - p[j * 64 + row]);
    float d = (row == j) ? 1.0f : 0.0f;
    Aug[row * 132 + j]      = d + s;
    Aug[row * 132 + 64 + j] = d - s;
  }
  __syncthreads();

  for (int k = 0; k < 64; ++k) {
    if (tid == 0) s_inv = 1.0f / Aug[k * 132 + k];
    __syncthreads();
    if (tid < 128) Aug[k * 132 + tid] *= s_inv;   // scale pivot row (both halves)
    __syncthreads();
    float factor = (row != k) ? Aug[row * 132 + k] : 0.0f;
    __syncthreads();
    if (row != k) {
      for (int jj = 0; jj < 32; ++jj) {
        int j = sub * 32 + jj;
        Aug[row * 132 + j] -= factor * Aug[k * 132 + j];
      }
    }
    __syncthreads();
  }

  float* dst = R + (size_t)z * 4096;
  for (int jj = 0; jj < 16; ++jj) {
    int j = sub * 16 + jj;
    float v = Aug[row * 132 + 64 + j];
    if (z < NUM_HEADS) dst[row * 64 + j] = v;   // R1: row-major
    else               dst[j * 64 + row] = v;   // R2: transposed (k-contiguous)
  }
}

// ---------------------------------------------------------------------------
// One 16x16 output tile of C = A(64x64) * B(64x64).  A row-major in LDS,
// B stored TRANSPOSED in LDS (BT[n][k]) so both fragments are k-contiguous
// per lane -> both merge into ds_load_2addr_b64.  16x V_WMMA_F32_16X16X4_F32.
// Fragment layouts per CDNA5 ISA 7.12.2:
//   A 16x4 : VGPR v, lane l -> A(row = l%16, k = v + 2*(l/16))
//   B 4x16 : VGPR v, lane l -> B(k = v + 2*(l/16), col = l%16)
//   C/D    : VGPR r, lane l -> C(row = r + 8*(l/16), col = l%16)
// ---------------------------------------------------------------------------
__device__ __forceinline__ v8f tile_gemm(const float* __restrict__ A,
                                         const float* __restrict__ BT,
                                         int tm, int tn, int l16, int lh) {
  v8f c = {};
#pragma unroll
  for (int k0 = 0; k0 < 64; k0 += 4) {
    const int ka = k0 + 2 * lh;
    v2f a, b;
    a.x = A [(tm * 16 + l16) * LDSTR + ka];
    a.y = A [(tm * 16 + l16) * LDSTR + ka + 1];
    b.x = BT[(tn * 16 + l16) * LDSTR + ka];
    b.y = BT[(tn * 16 + l16) * LDSTR + ka + 1];
    c = __builtin_amdgcn_wmma_f32_16x16x4_f32(false, a, false, b, (short)0, c,
                                              false, false);
  }
  return c;
}

// ---------------------------------------------------------------------------
// Kernel 2: out[b,c] = R1[c] * wkv[b,c] * R2[c].
// One block (8 wave32) per (b,c); each wave owns 2 of the 16 output tiles.
// GEMM1: T = W * R2   (R2 pre-transposed in ws; T written transposed to LDS,
//                      consecutive accumulator VGPRs -> 2x ds_store_b128/tile)
// GEMM2: O = R1 * T   (T^T is exactly the layout tile_gemm wants for B)
// ---------------------------------------------------------------------------
__global__ __launch_bounds__(256) void bilat_kernel(const float* __restrict__ wkv,
                                                    const float* __restrict__ R,
                                                    float* __restrict__ out) {
  __shared__ float sW  [64 * LDSTR];   // wkv tile (A of GEMM1), later T^T (B of GEMM2)
  __shared__ float sR1 [64 * LDSTR];   // R1 row-major (A of GEMM2)
  __shared__ float sR2T[64 * LDSTR];   // R2^T (B of GEMM1)

  const int bc = blockIdx.x;           // b*32 + c
  const int c  = bc & (NUM_HEADS - 1);
  const int tid = threadIdx.x;

  const float* gW  = wkv + (size_t)bc * 4096;
  const float* gR1 = R + (size_t)c * 4096;
  const float* gR2 = R + (size_t)(NUM_HEADS + c) * 4096;   // already transposed

  // Cooperative staging: 16B per lane per array, async direct-to-LDS.
  for (int i = tid; i < 1024; i += 256) {
    int r = i >> 4;
    int q = (i & 15) * 4;
    stage16(gW  + i * 4, &sW  [r * LDSTR + q]);
    stage16(gR1 + i * 4, &sR1 [r * LDSTR + q]);
    stage16(gR2 + i * 4, &sR2T[r * LDSTR + q]);
  }
  stage_join();
  __syncthreads();

  const int wave = tid >> 5;
  const int lane = tid & 31;
  const int l16  = lane & 15;
  const int lh   = lane >> 4;

  // GEMM1: T = W * R2 (tiles held in accumulators across the barrier)
  v8f acc[2];
#pragma unroll
  for (int i = 0; i < 2; ++i) {
    int t = wave + 8 * i;
    acc[i] = tile_gemm(sW, sR2T, t >> 2, t & 3, l16, lh);
  }
  __syncthreads();

  // Write T^T over sW: per lane the 8 accumulator elements land at
  // consecutive LDS addresses (16B aligned) -> merges to ds_store_b128.
#pragma unroll
  for (int i = 0; i < 2; ++i) {
    int t = wave + 8 * i;
    int tm = t >> 2, tn = t & 3;
#pragma unroll
    for (int r = 0; r < 8; ++r)
      sW[(tn * 16 + l16) * LDSTR + tm * 16 + 8 * lh + r] = acc[i][r];
  }
  __syncthreads();

  // GEMM2: O = R1 * T, stored straight to global (16-lane contiguous rows).
  float* gO = out + (size_t)bc * 4096;
#pragma unroll
  for (int i = 0; i < 2; ++i) {
    int t = wave + 8 * i;
    int tm = t >> 2, tn = t & 3;
    v8f o = tile_gemm(sR1, sW, tm, tn, l16, lh);
#pragma unroll
    for (int r = 0; r < 8; ++r)
      gO[(tm * 16 + r + 8 * lh) * 64 + tn * 16 + l16] = o[r];
  }
}

extern "C" void kernel_launch(void* const* d_in, const int* in_sizes, int n_in,
                              void* d_out, int out_size, void* d_ws, size_t ws_size,
                              hipStream_t stream) {
  const float* wkv     = (const float*)d_in[0];
  const float* p_left  = (const float*)d_in[1];
  const float* p_right = (const float*)d_in[2];
  float* out = (float*)d_out;
  float* R   = (float*)d_ws;   // 64 * 4096 floats = 1 MB: [R1(32) row-major | R2(32) transposed]

  cayley_kernel<<<64, 256, 0, stream>>>(p_left, p_right, R);
  bilat_kernel<<<BATCH * NUM_HEADS, 256, 0, stream>>>(wkv, R, out);
}